// BLCD_Loss_87076166960013
// MI455X (gfx1250) — compile-verified
//
#include <hip/hip_runtime.h>

typedef __attribute__((ext_vector_type(2))) float v2f;
typedef __attribute__((ext_vector_type(8))) float v8f;

#define NPT 256
#define DIM 256
#define KNBR 16
#define EPSF 1e-12f
#define MARGIN 0.6f
#define TTHRESH 0.0025f

__device__ __forceinline__ float wave_reduce_sum(float v) {
#pragma unroll
  for (int off = 16; off >= 1; off >>= 1)
    v += __shfl_xor(v, off, 32);
  return v;
}

// Kernel 1: per-row inverse L2 norms for both matrices (512 rows total),
// one wave (32 threads) per row. Also zero the scalar output.
__global__ void blcd_norms_kernel(const float* __restrict__ yi,
                                  const float* __restrict__ yt,
                                  float* __restrict__ rn,
                                  float* __restrict__ out) {
  const int row  = blockIdx.x;       // 0..511
  const int lane = threadIdx.x;      // 0..31
  const float* src = (row < NPT) ? yi : yt;
  const int r = row & (NPT - 1);
  float s = 0.f;
#pragma unroll
  for (int t = 0; t < DIM / 32; ++t) {
    float x = src[r * DIM + lane + 32 * t];
    s += x * x;
  }
  s = wave_reduce_sum(s);
  if (lane == 0) rn[row] = rsqrtf(s + EPSF);
  if (row == 0 && lane == 0) out[0] = 0.f;
}

// Kernel 2: one wave per 16x16 output tile.
// tile < 256  : G = dist(yi_n, yi_n)   -> Gdis
// tile >= 256 : C = dist(yit_n, yi_n)  -> Cdis
// Uses V_WMMA_F32_16X16X4_F32 over K=256 in steps of 4.
// Norms are folded into the epilogue: dot_n = dot_raw * rnA[i] * rnB[j].
__global__ void blcd_gram_wmma_kernel(const float* __restrict__ yi,
                                      const float* __restrict__ yt,
                                      const float* __restrict__ rn,
                                      float* __restrict__ Gdis,
                                      float* __restrict__ Cdis) {
  const int tile = blockIdx.x;            // 0..511
  const int m    = tile >> 8;             // 0 -> G, 1 -> C
  const int t    = tile & 255;
  const int i0   = (t >> 4) * 16;
  const int j0   = (t & 15) * 16;
  const int lane = threadIdx.x;           // 0..31
  const int half = lane >> 4;             // 0: K=k0,k0+1 ; 1: K=k0+2,k0+3
  const int r    = lane & 15;

  const float* Amat = m ? yt : yi;
  const float* rnA  = m ? (rn + NPT) : rn;
  const float* rnB  = rn;
  float* Out        = m ? Cdis : Gdis;

  const float* arow = Amat + (size_t)(i0 + r) * DIM;
  const float* brow = yi   + (size_t)(j0 + r) * DIM;

  v8f acc = {};
  for (int k0 = 0; k0 < DIM; k0 += 4) {
    const int kk = k0 + 2 * half;
    v2f a, b;
    a.x = arow[kk];
    a.y = arow[kk + 1];
    b.x = brow[kk];
    b.y = brow[kk + 1];
    // 8 args: (neg_a, A, neg_b, B, c_mod, C, reuse_a, reuse_b)
    acc = __builtin_amdgcn_wmma_f32_16x16x4_f32(
        false, a, false, b, (short)0, acc, false, false);
  }

  // Epilogue: C/D layout -> VGPR v holds M = v + 8*half, N = lane%16.
  const int gj = j0 + r;
  const float sb = rnB[gj];
#pragma unroll
  for (int v = 0; v < 8; ++v) {
    const int gi = i0 + v + 8 * half;
    const float dot = acc[v] * rnA[gi] * sb;
    const float ss  = fmaxf(2.0f - 2.0f * dot, 0.0f) + EPSF;
    Out[(size_t)gi * NPT + gj] = 0.5f * sqrtf(ss);
  }
}

// Kernel 3: one wave per row. 17 rounds of min-extraction over the row of
// Gdis (value + smallest-index tie-break, matching jax.lax.top_k ties).
// Round 0 is self; rounds 1..16 are the K neighbors (e1 terms); round 1's
// value is second_nn (e2 term). One atomicAdd per row into the scalar loss.
__global__ void blcd_topk_loss_kernel(const float* __restrict__ Gdis,
                                      const float* __restrict__ Cdis,
                                      float* __restrict__ out) {
  const int row  = blockIdx.x;    // 0..255
  const int lane = threadIdx.x;   // 0..31
  const float* g = Gdis + (size_t)row * NPT;

  float vals[8];
#pragma unroll
  for (int t = 0; t < 8; ++t) vals[t] = g[lane * 8 + t];

  unsigned used = 0u;
  float e1 = 0.f;
  float second_nn = 0.f;

  for (int s = 0; s <= KNBR; ++s) {
    float bv = __builtin_inff();
    int   bi = 0x7fffffff;
#pragma unroll
    for (int t = 0; t < 8; ++t) {
      if (!(used & (1u << t)) && vals[t] < bv) {
        bv = vals[t];
        bi = lane * 8 + t;
      }
    }
    // wave32 butterfly min-reduce with index tie-break (smaller index wins)
#pragma unroll
    for (int off = 16; off >= 1; off >>= 1) {
      float ov = __shfl_xor(bv, off, 32);
      int   oi = __shfl_xor(bi, off, 32);
      if (ov < bv || (ov == bv && oi < bi)) { bv = ov; bi = oi; }
    }
    if ((bi >> 3) == lane) used |= 1u << (bi & 7);   // owner marks consumed
    if (s >= 1) {
      if (lane == 0) {
        const float d2   = Cdis[(size_t)row * NPT + bi];
        const float diff = bv - d2;
        e1 += diff * diff;
      }
      if (s == 1) second_nn = bv;
    }
  }

  if (lane == 0) {
    const float diag = Cdis[(size_t)row * NPT + row];
    const float o    = diag + MARGIN - second_nn;
    const float e2   = fmaxf(o, 0.f);
    atomicAdd(out, e1 - (float)KNBR * TTHRESH + e2);
  }
}

extern "C" void kernel_launch(void* const* d_in, const int* in_sizes, int n_in,
                              void* d_out, int out_size, void* d_ws, size_t ws_size,
                              hipStream_t stream) {
  const float* yi = (const float*)d_in[0];
  const float* yt = (const float*)d_in[1];
  float* out = (float*)d_out;

  float* ws   = (float*)d_ws;
  float* rn   = ws;                       // 512 floats (rnorm yi rows, then yi_t rows)
  float* Gdis = ws + 512;                 // 256*256 floats
  float* Cdis = Gdis + NPT * NPT;         // 256*256 floats

  blcd_norms_kernel<<<512, 32, 0, stream>>>(yi, yt, rn, out);
  blcd_gram_wmma_kernel<<<512, 32, 0, stream>>>(yi, yt, rn, Gdis, Cdis);
  blcd_topk_loss_kernel<<<256, 32, 0, stream>>>(Gdis, Cdis, out);
}